// GCN_54150947668272
// MI455X (gfx1250) — compile-verified
//
#include <hip/hip_runtime.h>
#include <hip/hip_bf16.h>

typedef float v2f __attribute__((ext_vector_type(2)));
typedef float v8f __attribute__((ext_vector_type(8)));

#define F 128  // F_IN == HIDDEN == 128

// ---------------- utility kernels ----------------

__global__ void k_zero(float* __restrict__ p, long n) {
    long i = (long)blockIdx.x * blockDim.x + threadIdx.x;
    if (i < n) p[i] = 0.0f;
}

__global__ void k_deg(const int* __restrict__ dst, float* __restrict__ deg, int E) {
    int e = blockIdx.x * blockDim.x + threadIdx.x;
    if (e < E) atomicAdd(&deg[dst[e]], 1.0f);
}

__global__ void k_dinv(float* __restrict__ deg, int N) {
    int i = blockIdx.x * blockDim.x + threadIdx.x;
    if (i < N) deg[i] = rsqrtf(deg[i] + 1.0f);  // self-loop included
}

// ---------------- fp32 WMMA GEMM: H[nrows x 128] = X[nrows x 128] @ W[128 x 128] ----------------
// Block: 256 threads = 8 waves. Block computes a 16-row tile; wave w computes cols [16w,16w+16).
// K accumulated in steps of 4 via V_WMMA_F32_16X16X4_F32 (32 WMMAs / wave).

__global__ __launch_bounds__(256) void k_gemm_wmma(const float* __restrict__ X,
                                                   const float* __restrict__ W,
                                                   float* __restrict__ H, int nrows) {
    __shared__ float xs[16][F + 4];  // +4 pad: lane l reads row l, avoid 16-way bank conflicts
    const int tid = threadIdx.x;
    const int block_row = blockIdx.x * 16;
    const bool full_tile = (block_row + 16 <= nrows);

    // Stage the 16x128 A-tile (coalesced: 256 threads x 8 elements)
    if (full_tile) {
        #pragma unroll
        for (int j = 0; j < 8; ++j) {
            int i = tid + 256 * j;
            int r = i >> 7, c = i & (F - 1);
            xs[r][c] = X[(long)(block_row + r) * F + c];
        }
    } else {
        #pragma unroll
        for (int j = 0; j < 8; ++j) {
            int i = tid + 256 * j;
            int r = i >> 7, c = i & (F - 1);
            int row = block_row + r;
            xs[r][c] = (row < nrows) ? X[(long)row * F + c] : 0.0f;
        }
    }
    __syncthreads();

    const int wave = tid >> 5;          // N-tile select
    const int lane = tid & 31;
    const int half = lane >> 4;         // 0: K pair {0,1}; 1: K pair {2,3}
    const int l16  = lane & 15;
    const int col  = wave * 16 + l16;

    v8f c = {};
    #pragma unroll 4
    for (int k = 0; k < F; k += 4) {
        const int ka = k + half * 2;
        v2f a, b;
        // A 16x4 fragment: lanes 0-15 -> M=l16,K={k,k+1}; lanes 16-31 -> M=l16,K={k+2,k+3}
        a.x = xs[l16][ka];
        a.y = xs[l16][ka + 1];
        // B 4x16 fragment: N = col, K split across lane halves like A
        b.x = W[ka * F + col];
        b.y = W[(ka + 1) * F + col];
        c = __builtin_amdgcn_wmma_f32_16x16x4_f32(false, a, false, b,
                                                  (short)0, c, false, false);
    }

    // C/D layout: VGPR r -> M = r + 8*half, N = l16
    const int mbase = block_row + half * 8;
    if (full_tile) {
        // single base address, 8 stores with immediate offsets (r * 512B)
        float* __restrict__ p = H + (long)mbase * F + col;
        #pragma unroll
        for (int r = 0; r < 8; ++r) p[(long)r * F] = c[r];
    } else {
        #pragma unroll
        for (int r = 0; r < 8; ++r) {
            int row = mbase + r;
            if (row < nrows) H[(long)row * F + col] = c[r];
        }
    }
}

// ---------------- edge scatter: agg[dst] += dinv[src]*dinv[dst] * h[src] ----------------
// One wave per edge; each lane handles a float4 slice of the 128 features.
// Node feature buffers (25.6 MB) are L2-resident on the 192MB L2, so this is
// atomic-throughput bound, not HBM bound.

__global__ __launch_bounds__(256) void k_scatter(const float* __restrict__ h,
                                                 const int* __restrict__ src,
                                                 const int* __restrict__ dst,
                                                 const float* __restrict__ dinv,
                                                 float* __restrict__ agg, int E) {
    int e = (int)(((long)blockIdx.x * blockDim.x + threadIdx.x) >> 5);
    int lane = threadIdx.x & 31;
    if (e >= E) return;
    int s = src[e], d = dst[e];
    float norm = dinv[s] * dinv[d];
    float4 v = ((const float4*)(h + (long)s * F))[lane];
    float* ad = agg + (long)d * F + lane * 4;
    atomicAdd(ad + 0, norm * v.x);
    atomicAdd(ad + 1, norm * v.y);
    atomicAdd(ad + 2, norm * v.z);
    atomicAdd(ad + 3, norm * v.w);
}

// ---------------- self-loop + bias + ReLU (in place on agg) ----------------

__global__ void k_selfbias_relu(float* __restrict__ agg, const float* __restrict__ h,
                                const float* __restrict__ dinv, const float* __restrict__ b,
                                long n) {
    long i = (long)blockIdx.x * blockDim.x + threadIdx.x;
    if (i >= n) return;
    int node = (int)(i >> 7);
    int f = (int)(i & (F - 1));
    float di = dinv[node];
    float v = agg[i] + di * di * h[i] + b[f];
    agg[i] = fmaxf(v, 0.0f);
}

// ---------------- global mean pool accumulation ----------------

__global__ __launch_bounds__(256) void k_pool(const float* __restrict__ h,
                                              const int* __restrict__ batch,
                                              float* __restrict__ psum,
                                              float* __restrict__ pcnt, int N) {
    int node = (int)(((long)blockIdx.x * blockDim.x + threadIdx.x) >> 5);
    int lane = threadIdx.x & 31;
    if (node >= N) return;
    int g = batch[node];
    float4 v = ((const float4*)(h + (long)node * F))[lane];
    float* ps = psum + (long)g * F + lane * 4;
    atomicAdd(ps + 0, v.x);
    atomicAdd(ps + 1, v.y);
    atomicAdd(ps + 2, v.z);
    atomicAdd(ps + 3, v.w);
    if (lane == 0) atomicAdd(&pcnt[g], 1.0f);
}

// ---------------- classifier head: out[g][c] = (psum[g]/cnt[g]) . Wl[:,c] + bl[c] ----------------

__global__ void k_final(const float* __restrict__ psum, const float* __restrict__ pcnt,
                        const float* __restrict__ Wl, const float* __restrict__ bl,
                        float* __restrict__ out, int G, int C) {
    int i = blockIdx.x * blockDim.x + threadIdx.x;
    if (i >= G * C) return;
    int g = i / C, c = i % C;
    float inv = 1.0f / fmaxf(pcnt[g], 1.0f);
    float acc = 0.0f;
    #pragma unroll 4
    for (int k = 0; k < F; ++k) acc += psum[(long)g * F + k] * Wl[k * C + c];
    out[i] = acc * inv + bl[c];
}

// ---------------- launcher ----------------

extern "C" void kernel_launch(void* const* d_in, const int* in_sizes, int n_in,
                              void* d_out, int out_size, void* d_ws, size_t ws_size,
                              hipStream_t stream) {
    const float* x  = (const float*)d_in[0];
    const int* ei   = (const int*)d_in[1];
    const int* batch= (const int*)d_in[2];
    const float* W1 = (const float*)d_in[3];
    const float* b1 = (const float*)d_in[4];
    const float* W2 = (const float*)d_in[5];
    const float* b2 = (const float*)d_in[6];
    const float* Wl = (const float*)d_in[7];
    const float* bl = (const float*)d_in[8];
    float* out = (float*)d_out;

    const int N = in_sizes[0] / F;      // 50000 nodes
    const int E = in_sizes[1] / 2;      // 800000 edges
    const int C = in_sizes[8];          // 10 classes
    const int G = out_size / C;         // 128 graphs
    const int* src = ei;
    const int* dst = ei + E;

    // workspace layout (floats)
    float* ws   = (float*)d_ws;
    float* dinv = ws;                                   // N (deg -> dinv in place)
    float* hbuf = ws + (((long)N + 255) & ~255L);       // N*F
    float* abuf = hbuf + (long)N * F;                   // N*F
    float* psum = abuf + (long)N * F;                   // G*F
    float* pcnt = psum + (long)G * F;                   // G

    const long NF = (long)N * F;
    const int TB = 256;
    const int gNF   = (int)((NF + TB - 1) / TB);
    const int gN    = (N + TB - 1) / TB;
    const int gE    = (E + TB - 1) / TB;
    const int gEw   = (E + 7) / 8;       // 8 waves / block, 1 wave / edge
    const int gNw   = (N + 7) / 8;       // 8 waves / block, 1 wave / node
    const int gGemm = (N + 15) / 16;

    // degrees -> dinv (shared by both layers)
    k_zero<<<gN, TB, 0, stream>>>(dinv, N);
    k_deg<<<gE, TB, 0, stream>>>(dst, dinv, E);
    k_dinv<<<gN, TB, 0, stream>>>(dinv, N);

    // layer 1: h = x@W1 ; agg = scatter + self + b1 ; relu
    k_gemm_wmma<<<gGemm, TB, 0, stream>>>(x, W1, hbuf, N);
    k_zero<<<gNF, TB, 0, stream>>>(abuf, NF);
    k_scatter<<<gEw, TB, 0, stream>>>(hbuf, src, dst, dinv, abuf, E);
    k_selfbias_relu<<<gNF, TB, 0, stream>>>(abuf, hbuf, dinv, b1, NF);

    // layer 2: h2 = abuf@W2 ; agg2 = scatter + self + b2 ; relu
    k_gemm_wmma<<<gGemm, TB, 0, stream>>>(abuf, W2, hbuf, N);
    k_zero<<<gNF, TB, 0, stream>>>(abuf, NF);
    k_scatter<<<gEw, TB, 0, stream>>>(hbuf, src, dst, dinv, abuf, E);
    k_selfbias_relu<<<gNF, TB, 0, stream>>>(abuf, hbuf, dinv, b2, NF);

    // mean pool + classifier
    k_zero<<<((long)G * F + G + TB - 1) / TB, TB, 0, stream>>>(psum, (long)G * F + G);
    k_pool<<<gNw, TB, 0, stream>>>(abuf, batch, psum, pcnt, N);
    k_final<<<(G * C + TB - 1) / TB, TB, 0, stream>>>(psum, pcnt, Wl, bl, out, G, C);
}